// Net_13597866459428
// MI455X (gfx1250) — compile-verified
//
#include <hip/hip_runtime.h>
#include <math.h>

#define N_NODES 100000
#define N_EDGES 3200000
#define F_IN    512
#define HID     64
#define NCLS    64
#define KSTEPS  10
#define ALPHA   0.1f
#define SCAN_T  1024

typedef float v2f __attribute__((ext_vector_type(2)));
typedef float v8f __attribute__((ext_vector_type(8)));

// ---------------------------------------------------------------------------
// Fused GEMM + bias (+ optional ReLU):  D[N x 64] = A[N x K] @ B[K x 64] + b
// One wave32 computes a 16x64 strip: 4 accumulators, 4 x V_WMMA_F32_16X16X4_F32
// per k-step sharing a single A operand.  A is a read-once stream -> NT loads
// (keep the 192MB L2 for the propagation working set) + global_prefetch_b8
// one line ahead.  B (<=128KB) stays RT / L2-resident across all blocks.
// ---------------------------------------------------------------------------
__global__ void gemm_wmma_f32(const float* __restrict__ A,
                              const float* __restrict__ B,
                              const float* __restrict__ bias,
                              float* __restrict__ D,
                              int K, int relu) {
  const int lane = threadIdx.x & 31;
  const int wave = threadIdx.x >> 5;
  const int lr   = lane & 15;     // M (for A) / N (for B,C)
  const int hi   = lane >> 4;     // K-pair selector
  const int tile = blockIdx.x * (blockDim.x >> 5) + wave;  // 16-row tile
  if (tile * 16 >= N_NODES) return;                        // wave-uniform exit
  const int m0 = tile * 16;

  v8f acc[4] = {};
  const float* ap = A + (size_t)(m0 + lr) * K + hi * 2;
  const float* bp = B + (size_t)(hi * 2) * 64 + lr;

#pragma unroll 2
  for (int k = 0; k < K; k += 4) {
    v2f a;
    a.x = __builtin_nontemporal_load(ap);
    a.y = __builtin_nontemporal_load(ap + 1);
    __builtin_prefetch(ap + 32, 0, 0);   // one 128B line ahead on the A stream
#pragma unroll
    for (int nt = 0; nt < 4; ++nt) {
      v2f b;
      b.x = bp[nt * 16];
      b.y = bp[nt * 16 + 64];
      acc[nt] = __builtin_amdgcn_wmma_f32_16x16x4_f32(
          false, a, false, b, (short)0, acc[nt], false, false);
    }
    ap += 4;
    bp += 4 * 64;
  }

#pragma unroll
  for (int nt = 0; nt < 4; ++nt) {
    const float bv = bias[nt * 16 + lr];
#pragma unroll
    for (int j = 0; j < 8; ++j) {
      float v = acc[nt][j] + bv;
      if (relu) v = fmaxf(v, 0.0f);
      D[(size_t)(m0 + hi * 8 + j) * 64 + nt * 16 + lr] = v;
    }
  }
}

// ---------------------------------------------------------------------------
// CSR-by-destination build (one-time; removes all atomics from the K-loop)
// ---------------------------------------------------------------------------
__global__ void zero_cnt_kernel(unsigned* __restrict__ cnt) {
  const int i = blockIdx.x * blockDim.x + threadIdx.x;
  if (i < N_NODES) cnt[i] = 0u;
}

__global__ void count_kernel(const int* __restrict__ col,
                             unsigned* __restrict__ cnt) {
  const int e = blockIdx.x * blockDim.x + threadIdx.x;
  if (e < N_EDGES) atomicAdd(&cnt[col[e]], 1u);
}

// dinv[i] = 1/sqrt(in_degree + 1 self loop)
__global__ void dinv_kernel(const unsigned* __restrict__ cnt,
                            float* __restrict__ dinv) {
  const int i = blockIdx.x * blockDim.x + threadIdx.x;
  if (i < N_NODES) dinv[i] = rsqrtf((float)cnt[i] + 1.0f);
}

// Single-workgroup exclusive scan of cnt[N] -> start[N+1] (LDS Hillis-Steele)
__global__ void scan_kernel(const unsigned* __restrict__ cnt,
                            unsigned* __restrict__ start,
                            unsigned* __restrict__ cursor) {
  __shared__ unsigned sdata[SCAN_T];
  __shared__ unsigned s_carry;
  const int tid = threadIdx.x;
  if (tid == 0) s_carry = 0u;
  __syncthreads();
  for (int base = 0; base < N_NODES; base += SCAN_T) {
    const int i = base + tid;
    const unsigned v = (i < N_NODES) ? cnt[i] : 0u;
    sdata[tid] = v;
    __syncthreads();
#pragma unroll
    for (int off = 1; off < SCAN_T; off <<= 1) {
      const unsigned t = (tid >= off) ? sdata[tid - off] : 0u;
      __syncthreads();
      sdata[tid] += t;
      __syncthreads();
    }
    const unsigned cbase = s_carry;          // stable: last update was sync'd
    if (i < N_NODES) {
      const unsigned excl = cbase + sdata[tid] - v;
      start[i]  = excl;
      cursor[i] = excl;
    }
    __syncthreads();                         // all reads of s_carry complete
    if (tid == 0) s_carry = cbase + sdata[SCAN_T - 1];
    __syncthreads();
  }
  if (tid == 0) start[N_NODES] = s_carry;
}

// Place each edge into its destination's CSR segment as a packed 8-byte
// record {src, weight} -> one b64 load per edge in the hot gather loop.
__global__ void csr_fill_kernel(const int* __restrict__ row,
                                const int* __restrict__ col,
                                const float* __restrict__ dinv,
                                unsigned* __restrict__ cursor,
                                int2* __restrict__ recs) {
  const int e = blockIdx.x * blockDim.x + threadIdx.x;
  if (e < N_EDGES) {
    const int r = row[e];
    const int c = col[e];
    const unsigned p = atomicAdd(&cursor[c], 1u);
    recs[p] = make_int2(r, __float_as_int(dinv[r] * dinv[c]));
  }
}

// ---------------------------------------------------------------------------
// Propagation step, atomic-free:
//   hout[i] = (1-a) * ( dinv[i]^2*hin[i] + sum_{e in CSR[i]} w_e*hin[src_e] )
//             + a * h0[i]
// 16 lanes per node (float4 each = full 64-ch row); edge records + h buffers
// are L2-resident (192 MB L2 >> 25.6 MB per h buffer + 25.6 MB records).
// ---------------------------------------------------------------------------
__global__ void gather_kernel(const unsigned* __restrict__ start,
                              const int2* __restrict__ recs,
                              const float* __restrict__ dinv,
                              const float* __restrict__ hin,
                              const float* __restrict__ h0,
                              float* __restrict__ hout) {
  const unsigned t = blockIdx.x * blockDim.x + threadIdx.x;
  if (t >= (unsigned)N_NODES * 16u) return;
  const unsigned i  = t >> 4;
  const unsigned cg = (t & 15u) * 4u;

  const float di = dinv[i];
  const float4 self = *reinterpret_cast<const float4*>(hin + (size_t)i * 64 + cg);
  const float sw = di * di;
  float4 acc;
  acc.x = sw * self.x;
  acc.y = sw * self.y;
  acc.z = sw * self.z;
  acc.w = sw * self.w;

  const unsigned p1 = start[i + 1];
#pragma unroll 2
  for (unsigned p = start[i]; p < p1; ++p) {
    const int2 rec = recs[p];                 // single b64: {src, weight}
    const float w = __int_as_float(rec.y);
    const float4 hv = *reinterpret_cast<const float4*>(hin + (size_t)rec.x * 64 + cg);
    acc.x += w * hv.x;
    acc.y += w * hv.y;
    acc.z += w * hv.z;
    acc.w += w * hv.w;
  }

  const float4 z = *reinterpret_cast<const float4*>(h0 + (size_t)i * 64 + cg);
  float4 o;
  o.x = (1.0f - ALPHA) * acc.x + ALPHA * z.x;
  o.y = (1.0f - ALPHA) * acc.y + ALPHA * z.y;
  o.z = (1.0f - ALPHA) * acc.z + ALPHA * z.z;
  o.w = (1.0f - ALPHA) * acc.w + ALPHA * z.w;
  *reinterpret_cast<float4*>(hout + (size_t)i * 64 + cg) = o;
}

// ---------------------------------------------------------------------------
// log_softmax over C=64: one wave32 per node, 2 channels per lane,
// wave32-wide shuffle reductions.
// ---------------------------------------------------------------------------
__global__ void logsoftmax_kernel(const float* __restrict__ h,
                                  float* __restrict__ out) {
  const int lane = threadIdx.x & 31;
  const int wave = threadIdx.x >> 5;
  const int node = blockIdx.x * 8 + wave;   // blockDim = 256 -> 8 waves
  if (node >= N_NODES) return;

  const float2 v = *reinterpret_cast<const float2*>(h + (size_t)node * 64 + lane * 2);

  float m = fmaxf(v.x, v.y);
#pragma unroll
  for (int off = 16; off > 0; off >>= 1) m = fmaxf(m, __shfl_xor(m, off));

  float s = expf(v.x - m) + expf(v.y - m);
#pragma unroll
  for (int off = 16; off > 0; off >>= 1) s += __shfl_xor(s, off);

  const float ls = logf(s) + m;
  float2 o;
  o.x = v.x - ls;
  o.y = v.y - ls;
  *reinterpret_cast<float2*>(out + (size_t)node * 64 + lane * 2) = o;
}

// ---------------------------------------------------------------------------
extern "C" void kernel_launch(void* const* d_in, const int* in_sizes, int n_in,
                              void* d_out, int out_size, void* d_ws, size_t ws_size,
                              hipStream_t stream) {
  (void)in_sizes; (void)n_in; (void)out_size; (void)ws_size;

  const float* x  = (const float*)d_in[0];
  const int*   ei = (const int*)d_in[1];        // [2, E] flat: row then col
  const float* w1 = (const float*)d_in[2];
  const float* b1 = (const float*)d_in[3];
  const float* w2 = (const float*)d_in[4];
  const float* b2 = (const float*)d_in[5];
  const int* row = ei;
  const int* col = ei + N_EDGES;

  // workspace layout (kept 16-byte aligned per sub-buffer)
  char* ws = (char*)d_ws;
  const size_t NH = (size_t)N_NODES * 64;
  float*    hA     = (float*)ws;     ws += NH * 4;                    // hidden + ping
  float*    h0     = (float*)ws;     ws += NH * 4;                    // teleport src
  float*    hB     = (float*)ws;     ws += NH * 4;                    // pong
  float*    dinv   = (float*)ws;     ws += (size_t)N_NODES * 4;
  unsigned* cnt    = (unsigned*)ws;  ws += (size_t)N_NODES * 4;
  unsigned* start  = (unsigned*)ws;  ws += (size_t)(N_NODES + 4) * 4; // padded: keeps recs 16B-aligned
  unsigned* cursor = (unsigned*)ws;  ws += (size_t)N_NODES * 4;
  int2*     recs   = (int2*)ws;                                        // E x {src, w}

  const int TB = 256;

  // ---- one-time normalization + CSR-by-destination build
  zero_cnt_kernel<<<(N_NODES + TB - 1) / TB, TB, 0, stream>>>(cnt);
  count_kernel<<<(N_EDGES + TB - 1) / TB, TB, 0, stream>>>(col, cnt);
  dinv_kernel<<<(N_NODES + TB - 1) / TB, TB, 0, stream>>>(cnt, dinv);
  scan_kernel<<<1, SCAN_T, 0, stream>>>(cnt, start, cursor);
  csr_fill_kernel<<<(N_EDGES + TB - 1) / TB, TB, 0, stream>>>(row, col, dinv,
                                                              cursor, recs);

  // ---- MLP: hA = relu(x@w1+b1); h0 = hA@w2+b2   (6250 row-tiles, 4 waves/blk)
  const int tiles = N_NODES / 16;                 // 6250
  const int wpb = 128 / 32;                       // 4 waves per block
  gemm_wmma_f32<<<(tiles + wpb - 1) / wpb, 128, 0, stream>>>(x, w1, b1, hA, F_IN, 1);
  gemm_wmma_f32<<<(tiles + wpb - 1) / wpb, 128, 0, stream>>>(hA, w2, b2, h0, HID, 0);

  // ---- K atomic-free propagation steps, ping-pong hA/hB
  const float* hin = h0;
  float* bufs[2] = {hA, hB};
  const unsigned gthreads = (unsigned)N_NODES * 16u;
  for (int k = 0; k < KSTEPS; ++k) {
    float* hout = bufs[k & 1];
    gather_kernel<<<(gthreads + TB - 1) / TB, TB, 0, stream>>>(start, recs, dinv,
                                                               hin, h0, hout);
    hin = hout;
  }

  logsoftmax_kernel<<<(N_NODES + 7) / 8, TB, 0, stream>>>(hin, (float*)d_out);
}